// SPDNet_12893491822888
// MI455X (gfx1250) — compile-verified
//
#include <hip/hip_runtime.h>
#include <hip/hip_bf16.h>

typedef float v2f __attribute__((ext_vector_type(2)));
typedef float v8f __attribute__((ext_vector_type(8)));

#define LD 33            // row stride of padded 32x32 LDS matrices (gcd(33,64)=1)
#define MATF (32 * LD)   // floats per padded matrix
#define EPS_F 1e-4f

// ---------------------------------------------------------------------------
// One 16x16 output tile of C = A @ B via V_WMMA_F32_16X16X4_F32, K in steps of 4.
//   A: row-major in LDS, element A[m][k] at A[m*LD + k]
//   B: generic strides, element B[k][n] at B[k*ldbk + n*ldbn]  (transpose = swap)
// Fragment layouts per CDNA5 ISA 7.12.2 (wave32):
//   A 16x4 : lanes 0-15 M=lane (K=0,1 in v[0],v[1]); lanes 16-31 same M, K=2,3
//   B 4x16 : lanes 0-15 N=lane, VGPR r = row r; lanes 16-31 row r+2
//   C 16x16: VGPR r = row (r + 8*(lane>>4)), col = lane&15
// ---------------------------------------------------------------------------
__device__ __forceinline__ v8f wmma_gemm_tile(const float* A,
                                              const float* B, int ldbk, int ldbn,
                                              int mt, int nt, int K, int lane) {
  v8f c = {0.f, 0.f, 0.f, 0.f, 0.f, 0.f, 0.f, 0.f};
  const int m  = (lane & 15) + mt * 16;
  const int n  = (lane & 15) + nt * 16;
  const int kb = (lane >> 4) * 2;
  for (int k0 = 0; k0 < K; k0 += 4) {
    v2f a, b;
    a.x = A[m * LD + (k0 + kb)];
    a.y = A[m * LD + (k0 + kb + 1)];
    b.x = B[(k0 + kb) * ldbk + n * ldbn];
    b.y = B[(k0 + kb + 1) * ldbk + n * ldbn];
    c = __builtin_amdgcn_wmma_f32_16x16x4_f32(false, a, false, b, (short)0, c,
                                              false, false);
  }
  return c;
}

__device__ __forceinline__ void store_tile(float* C, int mt, int nt, v8f c, int lane) {
  const int hi = lane >> 4;
  const int n  = (lane & 15) + nt * 16;
#pragma unroll
  for (int r = 0; r < 8; ++r)
    C[(mt * 16 + r + 8 * hi) * LD + n] = c[r];
}

// ---------------------------------------------------------------------------
// Wave-cooperative cyclic Jacobi eigensolver for symmetric n x n (n <= 32).
// A (LDS, padded 32x33) is reduced toward diagonal; V accumulates eigenvectors
// (columns). Rotation scalars are uniform LDS broadcast reads; lane = row/col.
// Single-wave workgroup: __syncthreads() is a compiler fence (HW S_NOP).
// ---------------------------------------------------------------------------
__device__ void jacobi_sym(float* A, float* V, int n, int sweeps, int lane) {
  for (int idx = lane; idx < MATF; idx += 32) V[idx] = 0.f;
  __syncthreads();
  if (lane < n) V[lane * LD + lane] = 1.f;
  __syncthreads();
  for (int s = 0; s < sweeps; ++s) {
    for (int p = 0; p < n - 1; ++p) {
      for (int q = p + 1; q < n; ++q) {
        const float app = A[p * LD + p];
        const float aqq = A[q * LD + q];
        const float apq = A[p * LD + q];
        float c, sn;
        if (__builtin_fabsf(apq) > 1e-30f) {            // uniform branch
          const float tau = (aqq - app) / (2.f * apq);
          const float t   = __builtin_copysignf(1.f, tau) /
                          (__builtin_fabsf(tau) + __builtin_sqrtf(1.f + tau * tau));
          c  = 1.f / __builtin_sqrtf(1.f + t * t);
          sn = t * c;
        } else { c = 1.f; sn = 0.f; }
        if (lane < n) {                                  // rows p,q: J^T A
          const int j = lane;
          const float apj = A[p * LD + j], aqj = A[q * LD + j];
          A[p * LD + j] = c * apj - sn * aqj;
          A[q * LD + j] = sn * apj + c * aqj;
        }
        __syncthreads();
        if (lane < n) {                                  // cols p,q: (.)J ; V = V J
          const int i = lane;
          const float aip = A[i * LD + p], aiq = A[i * LD + q];
          A[i * LD + p] = c * aip - sn * aiq;
          A[i * LD + q] = sn * aip + c * aiq;
          const float vip = V[i * LD + p], viq = V[i * LD + q];
          V[i * LD + p] = c * vip - sn * viq;
          V[i * LD + q] = sn * vip + c * viq;
        }
        __syncthreads();
      }
    }
  }
}

// ---------------------------------------------------------------------------
// SPDNet forward: bimap(W1) -> reeig -> bimap(W2) -> (reeig+logeig fused, one
// eigh) -> upper-triangular. One wave32 workgroup per batch matrix.
// ---------------------------------------------------------------------------
__global__ void __launch_bounds__(32)
spdnet_kernel(const float* __restrict__ Xg, const float* __restrict__ W1g,
              const float* __restrict__ W2g, float* __restrict__ outg, int batch) {
  __shared__ float Xs[MATF], W1s[MATF], W2s[MATF],
                   Ta[MATF], Ma[MATF], Va[MATF], Ua[MATF];
  const int lane = threadIdx.x;

  // Zero-fill pads; stage W1 (20x30) and W2 (15x20) once per block.
  for (int idx = lane; idx < MATF; idx += 32) {
    W1s[idx] = 0.f; W2s[idx] = 0.f; Xs[idx] = 0.f; Ta[idx] = 0.f;
  }
  __syncthreads();
  for (int idx = lane; idx < 20 * 30; idx += 32)
    W1s[(idx / 30) * LD + idx % 30] = W1g[idx];
  for (int idx = lane; idx < 15 * 20; idx += 32)
    W2s[(idx / 20) * LD + idx % 20] = W2g[idx];
  __syncthreads();

  for (int b = blockIdx.x; b < batch; b += gridDim.x) {
    // X (30x30), zero-padded to 32x32
    for (int idx = lane; idx < 30 * 30; idx += 32)
      Xs[(idx / 30) * LD + idx % 30] = Xg[(size_t)b * 900 + idx];
    __syncthreads();

    // T = W1 @ X   (20x30, K=30 padded to 32)
    for (int mt = 0; mt < 2; ++mt)
      for (int nt = 0; nt < 2; ++nt)
        store_tile(Ta, mt, nt, wmma_gemm_tile(W1s, Xs, LD, 1, mt, nt, 32, lane), lane);
    __syncthreads();
    // M1 = T @ W1^T  (20x20): B[k][n] = W1[n][k]  -> strides (1, LD)
    for (int mt = 0; mt < 2; ++mt)
      for (int nt = 0; nt < 2; ++nt)
        store_tile(Ua, mt, nt, wmma_gemm_tile(Ta, W1s, 1, LD, mt, nt, 32, lane), lane);
    __syncthreads();
    // Ma = 0.5*(M1 + M1^T) + eps*I, zero outside 20x20
    for (int idx = lane; idx < MATF; idx += 32) {
      const int i = idx / LD, j = idx % LD;
      float v = 0.f;
      if (i < 20 && j < 20) {
        v = 0.5f * (Ua[i * LD + j] + Ua[j * LD + i]);
        if (i == j) v += EPS_F;
      }
      Ma[idx] = v;
    }
    __syncthreads();

    jacobi_sym(Ma, Va, 20, 7, lane);   // w = diag(Ma), V = Va

    // U = V * diag(max(w, eps)), zero-padded
    for (int idx = lane; idx < MATF; idx += 32) {
      const int i = idx / LD, j = idx % LD;
      float v = 0.f;
      if (i < 20 && j < 20)
        v = Va[i * LD + j] * fmaxf(Ma[j * LD + j], EPS_F);
      Ua[idx] = v;
    }
    __syncthreads();
    // A2 = U @ V^T  (20x20, K=20) -> reuse Xs
    for (int mt = 0; mt < 2; ++mt)
      for (int nt = 0; nt < 2; ++nt)
        store_tile(Xs, mt, nt, wmma_gemm_tile(Ua, Va, 1, LD, mt, nt, 20, lane), lane);
    __syncthreads();
    // T2 = W2 @ A2  (15x20, K=20)
    for (int nt = 0; nt < 2; ++nt)
      store_tile(Ta, 0, nt, wmma_gemm_tile(W2s, Xs, LD, 1, 0, nt, 20, lane), lane);
    __syncthreads();
    // M2 = T2 @ W2^T  (15x15, K=20)
    store_tile(Ua, 0, 0, wmma_gemm_tile(Ta, W2s, 1, LD, 0, 0, 20, lane), lane);
    __syncthreads();
    // Ma = 0.5*(M2 + M2^T) + eps*I, zero outside 15x15
    for (int idx = lane; idx < MATF; idx += 32) {
      const int i = idx / LD, j = idx % LD;
      float v = 0.f;
      if (i < 15 && j < 15) {
        v = 0.5f * (Ua[i * LD + j] + Ua[j * LD + i]);
        if (i == j) v += EPS_F;
      }
      Ma[idx] = v;
    }
    __syncthreads();

    jacobi_sym(Ma, Va, 15, 7, lane);   // fused reeig+logeig: one eigh

    // U2 = V2 * diag(log(max(w, eps)))
    for (int idx = lane; idx < MATF; idx += 32) {
      const int i = idx / LD, j = idx % LD;
      float v = 0.f;
      if (i < 15 && j < 15)
        v = Va[i * LD + j] * __logf(fmaxf(Ma[j * LD + j], EPS_F));
      Ua[idx] = v;
    }
    __syncthreads();

    // R = U2 @ V2^T (15x15, K=16); scatter upper-tri straight from D fragment.
    const v8f r = wmma_gemm_tile(Ua, Va, 1, LD, 0, 0, 16, lane);
    {
      const int hi = lane >> 4;
      const int j  = lane & 15;
#pragma unroll
      for (int rr = 0; rr < 8; ++rr) {
        const int i = rr + 8 * hi;                 // C/D layout: VGPR rr -> row
        if (i < 15 && j < 15 && j >= i) {
          const int off = i * 15 - (i * (i - 1)) / 2 + (j - i);
          outg[(size_t)b * 120 + off] = r[rr];
        }
      }
    }
    __syncthreads();   // Xs is overwritten next iteration
  }
}

extern "C" void kernel_launch(void* const* d_in, const int* in_sizes, int n_in,
                              void* d_out, int out_size, void* d_ws, size_t ws_size,
                              hipStream_t stream) {
  (void)n_in; (void)out_size; (void)d_ws; (void)ws_size;
  const float* X  = (const float*)d_in[0];   // (B, 30, 30)
  const float* W1 = (const float*)d_in[1];   // (20, 30)
  const float* W2 = (const float*)d_in[2];   // (15, 20)
  float* out = (float*)d_out;                // (B, 120)
  const int batch = in_sizes[0] / 900;
  const int blocks = (batch < 8192) ? batch : 8192;
  spdnet_kernel<<<dim3(blocks), dim3(32), 0, stream>>>(X, W1, W2, out, batch);
}